// KGAT_60035052863931
// MI455X (gfx1250) — compile-verified
//
#include <hip/hip_runtime.h>
#include <hip/hip_bf16.h>
#include <math.h>

#define N_USER   60000
#define N_ENT    140000
#define NNODE    200000
#define RREL     6
#define EDG      250000
#define DIM      64
#define NLAYER   3

typedef __attribute__((ext_vector_type(2))) float v2f;
typedef __attribute__((ext_vector_type(8))) float v8f;

static __device__ __forceinline__ v8f wmma4(v2f a, v2f b, v8f c) {
  // D = A(16x4,f32) * B(4x16,f32) + C(16x16,f32)
  return __builtin_amdgcn_wmma_f32_16x16x4_f32(false, a, false, b, (short)0, c,
                                               false, false);
}

static __device__ __forceinline__ void atomicMaxF(float* addr, float v) {
  // classic signed/unsigned trick; m initialized to -inf
  if (v >= 0.0f) {
    atomicMax((int*)addr, __float_as_int(v));
  } else {
    atomicMin((unsigned int*)addr, __float_as_uint(v));
  }
}

// ---------------- concat embeddings + write out columns [0:64) ----------------
__global__ void k_concat(const float* __restrict__ eu, const float* __restrict__ ee,
                         float* __restrict__ x0, float* __restrict__ out) {
  int t = blockIdx.x * blockDim.x + threadIdx.x;
  if (t >= NNODE * DIM) return;
  int node = t >> 6, d = t & 63;
  float v = (node < N_USER) ? eu[t] : ee[t - N_USER * DIM];
  x0[t] = v;
  out[(size_t)node * 256 + d] = v;
}

__global__ void k_initm(float* __restrict__ m) {
  int t = blockIdx.x * blockDim.x + threadIdx.x;
  if (t < NNODE) m[t] = -INFINITY;
}

// ---------------- edge attention scores: pai + segment max -------------------
__global__ __launch_bounds__(128)
void k_edge(const float* __restrict__ x, const int* __restrict__ edges,
            const float* __restrict__ transE, const float* __restrict__ rel,
            float* __restrict__ pai, float* __restrict__ m) {
  // B fragments of transE[r], pre-swizzled per lane: entry (k*4+n)*32+lane
  __shared__ v2f   sTb[2048];        // 16 KB
  __shared__ float sRel[64];
  __shared__ float sH[4][16 * 68];   // head rows per wave, stride 68 (bank-safe)
  __shared__ float sQ[4][16 * 68];   // tail rows per wave

  const int tid  = threadIdx.x;
  const int w    = tid >> 5;
  const int lane = tid & 31;
  const int l16  = lane & 15;
  const int half = lane >> 4;
  const int r    = blockIdx.y;

  // build swizzled B: {T[kk][col], T[kk+1][col]}, kk=4k+2*(ln>=16), col=16n+ln%16
  for (int i = tid; i < 2048; i += 128) {
    int k   = i >> 7;
    int n   = (i >> 5) & 3;
    int ln  = i & 31;
    int kk  = 4 * k + 2 * (ln >> 4);
    int col = n * 16 + (ln & 15);
    v2f b;
    b.x = transE[r * 4096 + kk * 64 + col];
    b.y = transE[r * 4096 + (kk + 1) * 64 + col];
    sTb[i] = b;
  }
  if (tid < 64) sRel[tid] = rel[r * 64 + tid];

  int tile = blockIdx.x * 4 + w;
  const bool valid = tile < (EDG / 16);
  if (!valid) tile = 0;
  const int e0 = tile * 16;

  int hidx = 0, qidx = 0;
  if (lane < 16) {
    const int* ep = edges + ((size_t)r * EDG + e0 + lane) * 2;
    hidx = ep[0];
    qidx = ep[1];
  }

  // stage 16 head rows + 16 tail rows (float4, coalesced; rows live in L2)
#pragma unroll
  for (int i = 0; i < 8; ++i) {
    int idx = lane + 32 * i;
    int row = idx >> 4, c4 = idx & 15;
    int h = __shfl(hidx, row);
    int q = __shfl(qidx, row);
    float4 hv = *((const float4*)(x + (size_t)h * 64) + c4);
    float4 qv = *((const float4*)(x + (size_t)q * 64) + c4);
    *((float4*)&sH[w][row * 68 + c4 * 4]) = hv;
    *((float4*)&sQ[w][row * 68 + c4 * 4]) = qv;
  }
  __syncthreads();

  v8f hr[4] = {};
  v8f tr[4] = {};
#pragma unroll
  for (int k = 0; k < 16; ++k) {
    const int krow = 4 * k + 2 * half;     // this lane's K pair
    v2f aH = *(const v2f*)&sH[w][l16 * 68 + krow];
    v2f aQ = *(const v2f*)&sQ[w][l16 * 68 + krow];
    const v2f* bp = &sTb[k * 128 + lane];
#pragma unroll
    for (int n = 0; n < 4; ++n) {
      v2f b = bp[n * 32];
      hr[n] = wmma4(aH, b, hr[n]);
      tr[n] = wmma4(aQ, b, tr[n]);
    }
  }

  // pai[M] = sum_f tr[M][f] * tanh(hr[M][f] + rel[f]); lane element: M=v+8*half, f=16n+l16
  float part[8];
#pragma unroll
  for (int v = 0; v < 8; ++v) part[v] = 0.0f;
#pragma unroll
  for (int n = 0; n < 4; ++n) {
    float rl = sRel[n * 16 + l16];
#pragma unroll
    for (int v = 0; v < 8; ++v) {
      float h = hr[n][v] + rl;
      part[v] += tr[n][v] * tanhf(h);
    }
  }
#pragma unroll
  for (int off = 1; off < 16; off <<= 1) {
#pragma unroll
    for (int v = 0; v < 8; ++v) part[v] += __shfl_xor(part[v], off);
  }

#pragma unroll
  for (int v = 0; v < 8; ++v) {
    int src = 8 * half + v;
    int hv = __shfl(hidx, src);      // head node of edge e0+src (uniform shfl)
    if (l16 == 0 && valid) {
      int e = e0 + src;
      pai[(size_t)r * EDG + e] = part[v];
      atomicMaxF(&m[hv], part[v]);
    }
  }
}

// ---------------- ex = exp(pai - m[row]), denom += ex ------------------------
__global__ void k_exp(float* __restrict__ pai, const float* __restrict__ m,
                      const int* __restrict__ edges, float* __restrict__ denom) {
  int e = blockIdx.x * blockDim.x + threadIdx.x;
  if (e >= RREL * EDG) return;
  int row = edges[(size_t)e * 2];
  float v = expf(pai[e] - m[row]);
  pai[e] = v;
  atomicAdd(&denom[row], v);
}

// ---------------- SpMM: nei[row] += (ex/denom[row]) * x[col] -----------------
__global__ void k_spmm(const float* __restrict__ ex, const float* __restrict__ denom,
                       const int* __restrict__ edges, const float* __restrict__ x,
                       float* __restrict__ nei) {
  int t = blockIdx.x * blockDim.x + threadIdx.x;
  if (t >= RREL * EDG * 32) return;
  int e = t >> 5;
  int d = (t & 31) * 2;
  int row = edges[(size_t)e * 2];
  int col = edges[(size_t)e * 2 + 1];
  float val = ex[e] / denom[row];
  const float* xp = x + (size_t)col * 64 + d;
  float* np = nei + (size_t)row * 64 + d;
  atomicAdd(np,     val * xp[0]);
  atomicAdd(np + 1, val * xp[1]);
}

// -------- layer: x' = lrelu((nei+x)(W1+b1)) + lrelu((nei*x)(W2+b2)) ----------
__global__ __launch_bounds__(64)
void k_layer(const float* __restrict__ nei, const float* __restrict__ x,
             const float* __restrict__ W1, const float* __restrict__ b1,
             const float* __restrict__ W2, const float* __restrict__ b2,
             float* __restrict__ xn, float* __restrict__ out, int layer) {
  __shared__ v2f sW1b[2048];         // swizzled (W1+b1) fragments, 16 KB
  __shared__ v2f sW2b[2048];         // swizzled (W2+b2) fragments, 16 KB
  __shared__ float sA[2][16 * 68];   // nei + x
  __shared__ float sP[2][16 * 68];   // nei * x

  const int tid  = threadIdx.x;
  const int w    = tid >> 5;
  const int lane = tid & 31;
  const int l16  = lane & 15;
  const int half = lane >> 4;

  // build swizzled weight fragments; bias added to weight matrix (faithful)
  for (int i = tid; i < 2048; i += 64) {
    int k   = i >> 7;
    int n   = (i >> 5) & 3;
    int ln  = i & 31;
    int kk  = 4 * k + 2 * (ln >> 4);
    int col = n * 16 + (ln & 15);
    float bc1 = b1[col], bc2 = b2[col];
    v2f w1v, w2v;
    w1v.x = W1[kk * 64 + col] + bc1;
    w1v.y = W1[(kk + 1) * 64 + col] + bc1;
    w2v.x = W2[kk * 64 + col] + bc2;
    w2v.y = W2[(kk + 1) * 64 + col] + bc2;
    sW1b[i] = w1v;
    sW2b[i] = w2v;
  }

  int tile = blockIdx.x * 2 + w;
  const bool valid = tile < (NNODE / 16);
  if (!valid) tile = 0;
  const int node0 = tile * 16;

#pragma unroll
  for (int i = 0; i < 8; ++i) {
    int idx = lane + 32 * i;
    int row = idx >> 4, c4 = idx & 15;
    size_t g = ((size_t)(node0 + row)) * 64 + c4 * 4;
    float4 nv = *(const float4*)(nei + g);
    float4 xv = *(const float4*)(x + g);
    float4 av, pv;
    av.x = nv.x + xv.x; av.y = nv.y + xv.y; av.z = nv.z + xv.z; av.w = nv.w + xv.w;
    pv.x = nv.x * xv.x; pv.y = nv.y * xv.y; pv.z = nv.z * xv.z; pv.w = nv.w * xv.w;
    *((float4*)&sA[w][row * 68 + c4 * 4]) = av;
    *((float4*)&sP[w][row * 68 + c4 * 4]) = pv;
  }
  __syncthreads();

  v8f S[4] = {};
  v8f B[4] = {};
#pragma unroll
  for (int k = 0; k < 16; ++k) {
    const int krow = 4 * k + 2 * half;
    v2f aS = *(const v2f*)&sA[w][l16 * 68 + krow];
    v2f aP = *(const v2f*)&sP[w][l16 * 68 + krow];
    const v2f* b1p = &sW1b[k * 128 + lane];
    const v2f* b2p = &sW2b[k * 128 + lane];
#pragma unroll
    for (int n = 0; n < 4; ++n) {
      S[n] = wmma4(aS, b1p[n * 32], S[n]);
      B[n] = wmma4(aP, b2p[n * 32], B[n]);
    }
  }

  // leaky-relu(0.2), combine, accumulate squared norm per node
  v8f X[4];
  float part[8];
#pragma unroll
  for (int v = 0; v < 8; ++v) part[v] = 0.0f;
#pragma unroll
  for (int n = 0; n < 4; ++n) {
#pragma unroll
    for (int v = 0; v < 8; ++v) {
      float s = S[n][v]; s = (s > 0.0f) ? s : 0.2f * s;
      float b = B[n][v]; b = (b > 0.0f) ? b : 0.2f * b;
      float xv = s + b;
      X[n][v] = xv;
      part[v] += xv * xv;
    }
  }
#pragma unroll
  for (int off = 1; off < 16; off <<= 1) {
#pragma unroll
    for (int v = 0; v < 8; ++v) part[v] += __shfl_xor(part[v], off);
  }

#pragma unroll
  for (int v = 0; v < 8; ++v) {
    float scale = 1.0f / fmaxf(sqrtf(part[v]), 1e-12f);
    int node = node0 + 8 * half + v;
#pragma unroll
    for (int n = 0; n < 4; ++n) {
      int f = n * 16 + l16;
      if (valid) {
        float xv = X[n][v];
        xn[(size_t)node * 64 + f] = xv;                         // unnormalized -> next layer
        out[(size_t)node * 256 + (size_t)(layer + 1) * 64 + f] = xv * scale;
      }
    }
  }
}

extern "C" void kernel_launch(void* const* d_in, const int* in_sizes, int n_in,
                              void* d_out, int out_size, void* d_ws, size_t ws_size,
                              hipStream_t stream) {
  (void)in_sizes; (void)n_in; (void)out_size; (void)ws_size;
  const float* eu    = (const float*)d_in[0];
  const float* ee    = (const float*)d_in[1];
  const float* er    = (const float*)d_in[2];
  const float* tE    = (const float*)d_in[3];
  const float* W1    = (const float*)d_in[4];
  const float* b1    = (const float*)d_in[5];
  const float* W2    = (const float*)d_in[6];
  const float* b2    = (const float*)d_in[7];
  const int*   edges = (const int*)d_in[8];
  float* out = (float*)d_out;

  // workspace layout (floats): x | nei | pai/ex | m | denom  (~110 MB)
  float* ws    = (float*)d_ws;
  float* x     = ws;
  float* nei   = x   + (size_t)NNODE * 64;
  float* pai   = nei + (size_t)NNODE * 64;
  float* m     = pai + (size_t)RREL * EDG;
  float* denom = m   + NNODE;

  k_concat<<<(NNODE * DIM + 255) / 256, 256, 0, stream>>>(eu, ee, x, out);
  k_initm <<<(NNODE + 255) / 256, 256, 0, stream>>>(m);
  hipMemsetAsync(denom, 0, NNODE * sizeof(float), stream);

  dim3 ge((EDG / 16 + 3) / 4, RREL);
  k_edge<<<ge, 128, 0, stream>>>(x, edges, tE, er, pai, m);
  k_exp<<<(RREL * EDG + 255) / 256, 256, 0, stream>>>(pai, m, edges, denom);

  for (int k = 0; k < NLAYER; ++k) {
    hipMemsetAsync(nei, 0, (size_t)NNODE * 64 * sizeof(float), stream);
    k_spmm<<<(RREL * EDG * 32 + 255) / 256, 256, 0, stream>>>(pai, denom, edges, x, nei);
    // in-place x update is safe: each tile reads only its own 16 rows before writing them
    k_layer<<<(NNODE / 16 + 1) / 2, 64, 0, stream>>>(
        nei, x, W1 + (size_t)k * 4096, b1 + (size_t)k * 64,
        W2 + (size_t)k * 4096, b2 + (size_t)k * 64, x, out, k);
  }
}